// Transformer_41609643164176
// MI455X (gfx1250) — compile-verified
//
#include <hip/hip_runtime.h>

typedef __attribute__((ext_vector_type(2))) float v2f;
typedef __attribute__((ext_vector_type(8))) float v8f;
typedef unsigned int v4u __attribute__((ext_vector_type(4)));
typedef int v8i_t __attribute__((ext_vector_type(8)));
typedef int v4i_t __attribute__((ext_vector_type(4)));

#define NSEQ 4096
#define TT   36     // tokens per sequence
#define TP   48     // padded to 3 x 16-token WMMA tiles
#define DD   4      // model dim
#define HH   4      // heads (head_dim = 1)
#define LL   8      // layers
#define FFD  2048   // FFN hidden
#define SPW  8      // sequences per workgroup (one per wave32)

__global__ __launch_bounds__(256)
void transformer_fused(const float* __restrict__ X,
                       const float* __restrict__ in_w,  const float* __restrict__ in_b,
                       const float* __restrict__ out_w, const float* __restrict__ out_b,
                       const float* __restrict__ ln1_w, const float* __restrict__ ln1_b,
                       const float* __restrict__ w1,    const float* __restrict__ b1,
                       const float* __restrict__ w2,    const float* __restrict__ b2,
                       const float* __restrict__ ln2_w, const float* __restrict__ ln2_b,
                       const float* __restrict__ cw1,   const float* __restrict__ cb1,
                       const float* __restrict__ cln1_w,const float* __restrict__ cln1_b,
                       const float* __restrict__ cw2,   const float* __restrict__ cb2,
                       const float* __restrict__ cln2_w,const float* __restrict__ cln2_b,
                       const float* __restrict__ cw3,   const float* __restrict__ cb3,
                       float* __restrict__ out)
{
    __shared__ __align__(16) float s_w2[DD][FFD];     // 32 KB: w2 (row-major), TDM-staged per layer
    __shared__ __align__(16) float s_x[SPW][TP][DD];  //  6 KB: activations (padded rows stay 0)
    __shared__ __align__(16) float s_qkv[SPW][TP][12];// 18 KB: q(0..3) k(4..7) v(8..11)
    __shared__ __align__(16) float s_o[SPW][TT][HH];  // 4.5 KB: attention output

    const int tid  = threadIdx.x;
    const int wave = tid >> 5;            // one sequence per wave
    const int lane = tid & 31;
    const int seq0 = blockIdx.x * SPW;

    // ---- init: zero s_x (padding rows must be 0) ----
    for (int i = tid; i < SPW*TP*DD; i += 256) ((float*)s_x)[i] = 0.f;
    __syncthreads();

    // ---- async global->LDS copy of X: one 16B token row per lane (ASYNCcnt path) ----
    for (int c = tid; c < SPW * TT; c += 256) {
        int s = c / TT, t0 = c - s * TT;
        unsigned ldsa = (unsigned)(size_t)(&s_x[s][t0][0]);
        unsigned long long ga =
            (unsigned long long)(size_t)(X + ((size_t)(seq0 + s) * TT + t0) * DD);
        asm volatile("global_load_async_to_lds_b128 %0, %1, off"
                     :: "v"(ldsa), "v"(ga) : "memory");
    }
    asm volatile("s_wait_asynccnt 0" ::: "memory");
    __syncthreads();

    // WMMA f32 16x16x4 lane decomposition (wave32):
    //   A: lane l holds row (l&15), K = {kb, kb+1} where kb = (l>>4)*2
    //   B: lane l holds col (l&15), K = {kb, kb+1}
    //   C: lane l holds col (l&15), rows rb8..rb8+7 in vgprs 0..7, rb8 = (l>>4)*8
    const int n   = lane & 15;
    const int kb  = (lane >> 4) << 1;
    const int rb8 = (lane >> 4) << 3;

    for (int l = 0; l < LL; ++l) {
        const float* w1l = w1 + (size_t)l * FFD * DD;
        const float* w2l = w2 + (size_t)l * DD * FFD;

        // ---- TDM: DMA this layer's w2 (4x2048 f32 = 32KB, flat 1-D tile) into LDS.
        // Issued by wave 0 only; overlapped with QKV/attention/LN1 below.
        if (tid < 32) {
            unsigned lds_dst = (unsigned)(size_t)(&s_w2[0][0]);
            unsigned long long ga = (unsigned long long)(size_t)w2l;
            v4u g0;
            g0[0] = 1u;                                          // count=1, user desc
            g0[1] = lds_dst;                                     // lds_addr
            g0[2] = (unsigned)(ga & 0xFFFFFFFFu);                // global_addr[31:0]
            g0[3] = (unsigned)((ga >> 32) & 0x01FFFFFFu)         // global_addr[56:32]
                    | (2u << 30);                                // type=2 (image)
            v8i_t g1;
            g1[0] = (2 << 16);                 // wg_mask=0, data_size=2 (4B), no flags
            g1[1] = (int)(8192u << 16);        // tensor_dim0[15:0]=8192 in bits[63:48]
            g1[2] = (1 << 16);                 // tensor_dim0 hi=0; tensor_dim1=1
            g1[3] = (int)(8192u << 16);        // tile_dim0=8192 in bits[127:112]
            g1[4] = 0;                         // tile_dim1=0 (unused), tile_dim2=0
            g1[5] = 8192;                      // tensor_dim0_stride[31:0]
            g1[6] = 0;
            g1[7] = 0;
            v4i_t z4 = {0, 0, 0, 0};
            v8i_t z8 = {0, 0, 0, 0, 0, 0, 0, 0};
            __builtin_amdgcn_tensor_load_to_lds(g0, g1, z4, z4, z8, 0);
        }

        // prefetch the w1 panel for this layer (global_prefetch_b8)
        for (int i = tid * 64; i < FFD*DD*4; i += 256 * 64)
            __builtin_prefetch((const char*)w1l + i, 0, 0);

        // ---- QKV projection: [16tok x 4] @ [4 x 12(pad16)] via v_wmma_f32_16x16x4_f32 ----
        const float* iwl = in_w + l * 12 * DD;
        v2f bq;
        bq.x = (n < 12) ? iwl[n*DD + kb]     : 0.f;
        bq.y = (n < 12) ? iwl[n*DD + kb + 1] : 0.f;
        const float qbias = (n < 12) ? in_b[l*12 + n] : 0.f;
        for (int t = 0; t < 3; ++t) {
            v2f a;
            a.x = s_x[wave][t*16 + n][kb];
            a.y = s_x[wave][t*16 + n][kb + 1];
            v8f c = {};
            c = __builtin_amdgcn_wmma_f32_16x16x4_f32(false, a, false, bq,
                                                      (short)0, c, false, false);
            if (n < 12) {
#pragma unroll
                for (int v = 0; v < 8; ++v)
                    s_qkv[wave][t*16 + rb8 + v][n] = c[v] + qbias;
            }
        }
        __syncthreads();

        // ---- causal attention, head_dim = 1, scale = 1 ----
        for (int idx = lane; idx < TT*HH; idx += 32) {
            int q = idx >> 2, h = idx & 3;
            float qv = s_qkv[wave][q][h];
            float m = -3.0e38f;
            for (int k = 0; k <= q; ++k)
                m = fmaxf(m, qv * s_qkv[wave][k][4 + h]);
            float sum = 0.f, o = 0.f;
            for (int k = 0; k <= q; ++k) {
                float e = __expf(qv * s_qkv[wave][k][4 + h] - m);
                sum += e;
                o   += e * s_qkv[wave][k][8 + h];
            }
            s_o[wave][q][h] = o / sum;
        }
        __syncthreads();

        // ---- out-proj + residual + LN1 ----
        const float* owl = out_w + l * DD * DD;
        for (int t = lane; t < TT; t += 32) {
            float y[DD];
#pragma unroll
            for (int d = 0; d < DD; ++d) {
                float acc = out_b[l*DD + d];
#pragma unroll
                for (int h = 0; h < HH; ++h) acc += s_o[wave][t][h] * owl[d*DD + h];
                y[d] = s_x[wave][t][d] + acc;
            }
            float mean = 0.25f * (y[0]+y[1]+y[2]+y[3]);
            float var = 0.f;
#pragma unroll
            for (int d = 0; d < DD; ++d) { float z = y[d]-mean; var += z*z; }
            float r = rsqrtf(var * 0.25f + 1e-5f);
#pragma unroll
            for (int d = 0; d < DD; ++d)
                s_x[wave][t][d] = (y[d]-mean)*r*ln1_w[l*DD+d] + ln1_b[l*DD+d];
        }

        // w2 DMA must have landed before the FFN touches s_w2
        if (tid < 32) __builtin_amdgcn_s_wait_tensorcnt(0);
        __syncthreads();

        // ---- FFN: WMMA over FF in 16-col chunks, K-reduce for ffn2 in C-layout ----
        const float* b1l = b1 + l * FFD;
        const float b20 = b2[l*DD+0], b21 = b2[l*DD+1], b22 = b2[l*DD+2], b23 = b2[l*DD+3];
        for (int t = 0; t < 3; ++t) {
            v2f a;
            a.x = s_x[wave][t*16 + n][kb];
            a.y = s_x[wave][t*16 + n][kb + 1];
            float acc[8][4];
#pragma unroll
            for (int v = 0; v < 8; ++v)
#pragma unroll
                for (int d = 0; d < 4; ++d) acc[v][d] = 0.f;

#pragma unroll 4
            for (int ff0 = 0; ff0 < FFD; ff0 += 16) {
                const float* wr = w1l + (size_t)(ff0 + n) * DD + kb;
                v2f b; b.x = wr[0]; b.y = wr[1];
                v8f c = {};
                c = __builtin_amdgcn_wmma_f32_16x16x4_f32(false, a, false, b,
                                                          (short)0, c, false, false);
                float hb  = b1l[ff0 + n];
                float w20 = s_w2[0][ff0 + n], w21 = s_w2[1][ff0 + n];
                float w22 = s_w2[2][ff0 + n], w23 = s_w2[3][ff0 + n];
#pragma unroll
                for (int v = 0; v < 8; ++v) {
                    float hv = fmaxf(c[v] + hb, 0.f);      // bias + relu
                    acc[v][0] += hv * w20;  acc[v][1] += hv * w21;
                    acc[v][2] += hv * w22;  acc[v][3] += hv * w23;
                }
            }
            // finish K-reduction across the 16 lanes of each half-wave
#pragma unroll
            for (int off = 1; off < 16; off <<= 1) {
#pragma unroll
                for (int v = 0; v < 8; ++v)
#pragma unroll
                    for (int d = 0; d < 4; ++d)
                        acc[v][d] += __shfl_xor(acc[v][d], off, 32);
            }
            if ((lane & 15) == 0) {    // lane 0 -> rows 0..7, lane 16 -> rows 8..15
#pragma unroll
                for (int v = 0; v < 8; ++v) {
                    int tok = t*16 + rb8 + v;
                    if (tok < TT) {
                        float y[4];
                        y[0] = s_x[wave][tok][0] + acc[v][0] + b20;
                        y[1] = s_x[wave][tok][1] + acc[v][1] + b21;
                        y[2] = s_x[wave][tok][2] + acc[v][2] + b22;
                        y[3] = s_x[wave][tok][3] + acc[v][3] + b23;
                        float mean = 0.25f * (y[0]+y[1]+y[2]+y[3]);
                        float var = 0.f;
#pragma unroll
                        for (int d = 0; d < 4; ++d) { float z = y[d]-mean; var += z*z; }
                        float r = rsqrtf(var * 0.25f + 1e-5f);
#pragma unroll
                        for (int d = 0; d < 4; ++d)
                            s_x[wave][tok][d] = (y[d]-mean)*r*ln2_w[l*DD+d] + ln2_b[l*DD+d];
                    }
                }
            }
        }
        __syncthreads();
    }

    // ---- classifier head ----
    const int seq = seq0 + wave;
    for (int t = lane; t < TT; t += 32) {
        float x0 = s_x[wave][t][0], x1 = s_x[wave][t][1];
        float x2 = s_x[wave][t][2], x3 = s_x[wave][t][3];
        float h1[16];
#pragma unroll
        for (int j = 0; j < 16; ++j)
            h1[j] = cb1[j] + x0*cw1[j*4] + x1*cw1[j*4+1] + x2*cw1[j*4+2] + x3*cw1[j*4+3];
        float m = 0.f;
#pragma unroll
        for (int j = 0; j < 16; ++j) m += h1[j];
        m *= (1.f/16.f);
        float var = 0.f;
#pragma unroll
        for (int j = 0; j < 16; ++j) { float z = h1[j]-m; var += z*z; }
        float r = rsqrtf(var*(1.f/16.f) + 1e-5f);
#pragma unroll
        for (int j = 0; j < 16; ++j)
            h1[j] = fmaxf((h1[j]-m)*r*cln1_w[j] + cln1_b[j], 0.f);

        float h2[4];
#pragma unroll
        for (int j = 0; j < 4; ++j) {
            float acc = cb2[j];
#pragma unroll
            for (int k = 0; k < 16; ++k) acc += h1[k]*cw2[j*16+k];
            h2[j] = acc;
        }
        float m2 = 0.25f*(h2[0]+h2[1]+h2[2]+h2[3]);
        float v2 = 0.f;
#pragma unroll
        for (int j = 0; j < 4; ++j) { float z = h2[j]-m2; v2 += z*z; }
        float r2 = rsqrtf(v2*0.25f + 1e-5f);
#pragma unroll
        for (int j = 0; j < 4; ++j)
            h2[j] = fmaxf((h2[j]-m2)*r2*cln2_w[j] + cln2_b[j], 0.f);

        float o0 = cb3[0] + h2[0]*cw3[0] + h2[1]*cw3[1] + h2[2]*cw3[2] + h2[3]*cw3[3];
        float o1 = cb3[1] + h2[0]*cw3[4] + h2[1]*cw3[5] + h2[2]*cw3[6] + h2[3]*cw3[7];
        size_t base = ((size_t)seq*TT + t)*2;
        out[base]     = o0;
        out[base + 1] = o1;
    }
}

extern "C" void kernel_launch(void* const* d_in, const int* in_sizes, int n_in,
                              void* d_out, int out_size, void* d_ws, size_t ws_size,
                              hipStream_t stream) {
    const float* X      = (const float*)d_in[0];
    const float* in_w   = (const float*)d_in[1];
    const float* in_b   = (const float*)d_in[2];
    const float* out_w  = (const float*)d_in[3];
    const float* out_b  = (const float*)d_in[4];
    const float* ln1_w  = (const float*)d_in[5];
    const float* ln1_b  = (const float*)d_in[6];
    const float* w1     = (const float*)d_in[7];
    const float* b1     = (const float*)d_in[8];
    const float* w2     = (const float*)d_in[9];
    const float* b2     = (const float*)d_in[10];
    const float* ln2_w  = (const float*)d_in[11];
    const float* ln2_b  = (const float*)d_in[12];
    const float* cw1    = (const float*)d_in[13];
    const float* cb1    = (const float*)d_in[14];
    const float* cln1_w = (const float*)d_in[15];
    const float* cln1_b = (const float*)d_in[16];
    const float* cw2    = (const float*)d_in[17];
    const float* cb2    = (const float*)d_in[18];
    const float* cln2_w = (const float*)d_in[19];
    const float* cln2_b = (const float*)d_in[20];
    const float* cw3    = (const float*)d_in[21];
    const float* cb3    = (const float*)d_in[22];
    float* out = (float*)d_out;

    dim3 grid(NSEQ / SPW);   // 512 workgroups, 8 wave32 each (1 sequence/wave)
    dim3 block(256);
    transformer_fused<<<grid, block, 0, stream>>>(
        X, in_w, in_b, out_w, out_b, ln1_w, ln1_b, w1, b1, w2, b2,
        ln2_w, ln2_b, cw1, cb1, cln1_w, cln1_b, cw2, cb2, cln2_w, cln2_b,
        cw3, cb3, out);
}